// GATEncoder_21663815041736
// MI455X (gfx1250) — compile-verified
//
#include <hip/hip_runtime.h>
#include <math.h>

typedef __attribute__((ext_vector_type(2))) float v2f;
typedef __attribute__((ext_vector_type(8))) float v8f;

#define HC 256   // heads * head_dim
#define NH 4     // heads

// ---------------------------------------------------------------------------
// Init: out[n][j] = bias[j]; amax = -inf; denom = 0; kvec = 0; sum_ea = 0
// ---------------------------------------------------------------------------
__global__ void k_init(float* __restrict__ out, float* __restrict__ amax,
                       float* __restrict__ denom, float* __restrict__ kvec,
                       float* __restrict__ sum_ea, const float* __restrict__ bias,
                       int N) {
    int idx = blockIdx.x * blockDim.x + threadIdx.x;
    if (idx < N * HC) out[idx] = bias[idx & (HC - 1)];
    if (idx < N * NH) { amax[idx] = -__builtin_inff(); denom[idx] = 0.0f; }
    if (idx < NH) kvec[idx] = 0.0f;
    if (idx == 0) *sum_ea = 0.0f;
}

// ---------------------------------------------------------------------------
// kvec[h] = sum_c W_edge[h*64+c] * att_edge[h*64+c]   (K=1 collapse of a_edge)
// ---------------------------------------------------------------------------
__global__ void k_prep(const float* __restrict__ W_edge,
                       const float* __restrict__ att_edge,
                       float* __restrict__ kvec) {
    int j = threadIdx.x;  // 256 threads
    atomicAdd(&kvec[j >> 6], W_edge[j] * att_edge[j]);
}

// ---------------------------------------------------------------------------
// sum of edge_attr (for self-loop fill value = mean)
// ---------------------------------------------------------------------------
__global__ void k_sum_ea(const float* __restrict__ ea, float* __restrict__ sum_ea,
                         int E) {
    int i = blockIdx.x * blockDim.x + threadIdx.x;
    float v = (i < E) ? ea[i] : 0.0f;
    #pragma unroll
    for (int off = 16; off > 0; off >>= 1) v += __shfl_xor(v, off, 32);
    if ((threadIdx.x & 31) == 0) atomicAdd(sum_ea, v);
}

// ---------------------------------------------------------------------------
// h = x @ W^T  via V_WMMA_F32_16X16X4_F32 (K=2 padded to 4, exact f32).
// One wave computes a 16-node x 256-channel stripe as 16 WMMA tiles.
// Branch-free operand setup: loads use (lane&15) addressing (in-bounds dupes)
// and lanes 16-31 are zeroed with value selects -> no EXEC save/restore, the
// 16 B-tile loads clause up front, WMMAs issue back-to-back.
// ---------------------------------------------------------------------------
__global__ void __launch_bounds__(256) k_proj_wmma(const float* __restrict__ x,
                                                   const float* __restrict__ W,
                                                   float* __restrict__ h, int N) {
    const int lane  = threadIdx.x & 31;
    const int half  = lane & 15;
    const bool lo   = lane < 16;
    const int wave  = blockIdx.x * (blockDim.x >> 5) + (threadIdx.x >> 5);
    const int m0    = wave * 16;
    if (m0 >= N) return;  // wave-uniform exit: EXEC all-1s for WMMA below

    const v2f* __restrict__ xv = (const v2f*)x;   // x is [N,2], 8B rows
    const v2f* __restrict__ Wv = (const v2f*)W;   // W is [256,2], 8B rows

    // A tile: lanes 0-15 hold rows m0..m0+15 (K=0,1); lanes 16-31 = K=2,3 = 0
    v2f a = xv[m0 + half];
    a.x = lo ? a.x : 0.0f;
    a.y = lo ? a.y : 0.0f;

    // Preload all 16 B tiles: B[k][col] = W[col][k]
    v2f btile[16];
    #pragma unroll
    for (int t = 0; t < 16; ++t) {
        v2f b = Wv[t * 16 + half];
        btile[t].x = lo ? b.x : 0.0f;
        btile[t].y = lo ? b.y : 0.0f;
    }

    // Single store base; all result stores are immediate offsets from it.
    const int row_base = m0 + (lo ? 0 : 8);
    float* __restrict__ hp = h + (size_t)row_base * HC + half;

    #pragma unroll
    for (int t = 0; t < 16; ++t) {
        v8f c = {};
        c = __builtin_amdgcn_wmma_f32_16x16x4_f32(
                /*neg_a=*/false, a, /*neg_b=*/false, btile[t],
                /*c_mod=*/(short)0, c, /*reuse_a=*/false, /*reuse_b=*/false);
        #pragma unroll
        for (int v = 0; v < 8; ++v)
            hp[t * 16 + v * HC] = c[v];   // D: VGPR v -> row (base+v), col half
    }
}

// ---------------------------------------------------------------------------
// a_src[n][h] = <h[n], att_src[h]> ; a_dst likewise. One wave per node,
// b128 loads; half-wave owns heads {i*2 + lane>>4}; 4-step shfl_xor reduce.
// ---------------------------------------------------------------------------
__global__ void __launch_bounds__(256) k_attn(const float* __restrict__ h,
                                              const float* __restrict__ att_src,
                                              const float* __restrict__ att_dst,
                                              float* __restrict__ a_src,
                                              float* __restrict__ a_dst, int N) {
    const int lane = threadIdx.x & 31;
    const int node = blockIdx.x * 8 + (threadIdx.x >> 5);
    if (node >= N) return;

    const float4* __restrict__ hp = (const float4*)(h + (size_t)node * HC);
    const float4* __restrict__ ap = (const float4*)att_src;
    const float4* __restrict__ bp = (const float4*)att_dst;

    float s[2], d[2];
    #pragma unroll
    for (int i = 0; i < 2; ++i) {
        int f = i * 32 + lane;                 // float4 idx; head = (f>>4)
        float4 hv = hp[f];
        float4 av = ap[f];
        float4 bv = bp[f];
        s[i] = hv.x * av.x + hv.y * av.y + hv.z * av.z + hv.w * av.w;
        d[i] = hv.x * bv.x + hv.y * bv.y + hv.z * bv.z + hv.w * bv.w;
    }
    // reduce within each 16-lane half (offsets < 16 stay in-half)
    #pragma unroll
    for (int off = 8; off > 0; off >>= 1) {
        #pragma unroll
        for (int i = 0; i < 2; ++i) {
            s[i] += __shfl_xor(s[i], off, 32);
            d[i] += __shfl_xor(d[i], off, 32);
        }
    }
    const int hb = lane >> 4;                  // 0: heads {0,2}; 1: heads {1,3}
    if ((lane & 15) == 0) {
        a_src[node * NH + hb]     = s[0];
        a_src[node * NH + 2 + hb] = s[1];
        a_dst[node * NH + hb]     = d[0];
        a_dst[node * NH + 2 + hb] = d[1];
    }
}

// ---------------------------------------------------------------------------
// Per-edge alpha = leaky_relu(a_src[s]+a_dst[d]+ea*k, 0.2); atomic segment max
// ---------------------------------------------------------------------------
__device__ __forceinline__ void atomic_max_f32(float* addr, float v) {
    if (v >= 0.0f) atomicMax((int*)addr, __float_as_int(v));
    else           atomicMin((unsigned int*)addr, (unsigned int)__float_as_int(v));
}

__global__ void k_edge_alpha(const long long* __restrict__ ei,
                             const float* __restrict__ edge_attr,
                             const float* __restrict__ a_src,
                             const float* __restrict__ a_dst,
                             const float* __restrict__ kvec,
                             const float* __restrict__ sum_ea,
                             float* __restrict__ alpha, float* __restrict__ amax,
                             int E, int N) {
    int e = blockIdx.x * blockDim.x + threadIdx.x;
    if (e >= E + N) return;
    int s, d; float ea;
    if (e < E) { s = (int)ei[e]; d = (int)ei[E + e]; ea = edge_attr[e]; }
    else       { s = d = e - E; ea = *sum_ea / (float)E; }

    const float4 as = *(const float4*)(a_src + (size_t)s * NH);
    const float4 ad = *(const float4*)(a_dst + (size_t)d * NH);
    const float4 kv = *(const float4*)kvec;

    float4 al;
    al.x = as.x + ad.x + ea * kv.x;
    al.y = as.y + ad.y + ea * kv.y;
    al.z = as.z + ad.z + ea * kv.z;
    al.w = as.w + ad.w + ea * kv.w;
    al.x = (al.x > 0.0f) ? al.x : 0.2f * al.x;
    al.y = (al.y > 0.0f) ? al.y : 0.2f * al.y;
    al.z = (al.z > 0.0f) ? al.z : 0.2f * al.z;
    al.w = (al.w > 0.0f) ? al.w : 0.2f * al.w;
    *(float4*)(alpha + (size_t)e * NH) = al;

    float* am = amax + (size_t)d * NH;
    atomic_max_f32(am + 0, al.x);
    atomic_max_f32(am + 1, al.y);
    atomic_max_f32(am + 2, al.z);
    atomic_max_f32(am + 3, al.w);
}

// ---------------------------------------------------------------------------
// ex = exp(alpha - amax[dst]); overwrite alpha with ex; denom[dst] += ex
// ---------------------------------------------------------------------------
__global__ void k_edge_exp(const long long* __restrict__ ei,
                           float* __restrict__ alpha,
                           const float* __restrict__ amax,
                           float* __restrict__ denom, int E, int N) {
    int e = blockIdx.x * blockDim.x + threadIdx.x;
    if (e >= E + N) return;
    int d = (e < E) ? (int)ei[E + e] : (e - E);

    const float4 al = *(const float4*)(alpha + (size_t)e * NH);
    const float4 am = *(const float4*)(amax + (size_t)d * NH);
    float4 ex;
    ex.x = __expf(al.x - am.x);
    ex.y = __expf(al.y - am.y);
    ex.z = __expf(al.z - am.z);
    ex.w = __expf(al.w - am.w);
    *(float4*)(alpha + (size_t)e * NH) = ex;

    float* dn = denom + (size_t)d * NH;
    atomicAdd(dn + 0, ex.x);
    atomicAdd(dn + 1, ex.y);
    atomicAdd(dn + 2, ex.z);
    atomicAdd(dn + 3, ex.w);
}

// ---------------------------------------------------------------------------
// Scatter: one wave per edge; 2 x b128 loads of h[src] per lane (L2-resident),
// 8 coalesced f32 atomic adds into out[dst]. head = (lane>>4) + 2*i.
// ---------------------------------------------------------------------------
__global__ void __launch_bounds__(256) k_scatter(const long long* __restrict__ ei,
                                                 const float* __restrict__ alpha,
                                                 const float* __restrict__ denom,
                                                 const float* __restrict__ h,
                                                 float* __restrict__ out,
                                                 int E, int N) {
    const int lane = threadIdx.x & 31;
    const int e = blockIdx.x * 8 + (threadIdx.x >> 5);
    if (e >= E + N) return;
    int s, d;
    if (e < E) { s = (int)ei[e]; d = (int)ei[E + e]; }
    else       { s = d = e - E; }

    const float4 ex = *(const float4*)(alpha + (size_t)e * NH);
    const float4 dn = *(const float4*)(denom + (size_t)d * NH);
    const float coef[NH] = { ex.x / dn.x, ex.y / dn.y, ex.z / dn.z, ex.w / dn.w };

    const float4* __restrict__ hs = (const float4*)(h + (size_t)s * HC);
    float*        __restrict__ od = out + (size_t)d * HC;
    #pragma unroll
    for (int i = 0; i < 2; ++i) {
        const int f  = i * 32 + lane;          // float4 idx; channels f*4..f*4+3
        const float4 hv = hs[f];
        const float cf  = coef[(lane >> 4) + i * 2];
        const int c0 = f * 4;
        atomicAdd(&od[c0 + 0], cf * hv.x);
        atomicAdd(&od[c0 + 1], cf * hv.y);
        atomicAdd(&od[c0 + 2], cf * hv.z);
        atomicAdd(&od[c0 + 3], cf * hv.w);
    }
}

// ---------------------------------------------------------------------------
extern "C" void kernel_launch(void* const* d_in, const int* in_sizes, int n_in,
                              void* d_out, int out_size, void* d_ws, size_t ws_size,
                              hipStream_t stream) {
    (void)n_in; (void)out_size; (void)ws_size;
    const float*     x        = (const float*)d_in[0];
    const long long* ei       = (const long long*)d_in[1];  // int64 [2,E]
    const float*     edge_att = (const float*)d_in[2];
    const float*     W        = (const float*)d_in[3];
    const float*     att_src  = (const float*)d_in[4];
    const float*     att_dst  = (const float*)d_in[5];
    const float*     W_edge   = (const float*)d_in[6];
    const float*     att_edge = (const float*)d_in[7];
    const float*     bias     = (const float*)d_in[8];
    float* out = (float*)d_out;

    const int N  = in_sizes[0] / 2;   // 50000
    const int E  = in_sizes[2];       // 800000
    const int ET = E + N;

    // workspace layout (all 16B-aligned: every chunk is a multiple of 4 floats)
    float* h      = (float*)d_ws;                 // N*256
    float* a_src  = h     + (size_t)N * HC;       // N*4
    float* a_dst  = a_src + (size_t)N * NH;       // N*4
    float* amax   = a_dst + (size_t)N * NH;       // N*4
    float* denom  = amax  + (size_t)N * NH;       // N*4
    float* alpha  = denom + (size_t)N * NH;       // (E+N)*4
    float* kvec   = alpha + (size_t)ET * NH;      // 4
    float* sum_ea = kvec + NH;                    // 1

    k_init<<<(N * HC + 255) / 256, 256, 0, stream>>>(out, amax, denom, kvec, sum_ea, bias, N);
    k_prep<<<1, HC, 0, stream>>>(W_edge, att_edge, kvec);
    k_sum_ea<<<(E + 255) / 256, 256, 0, stream>>>(edge_att, sum_ea, E);

    const int waves = (N + 15) / 16;
    k_proj_wmma<<<(waves + 7) / 8, 256, 0, stream>>>(x, W, h, N);

    k_attn<<<(N + 7) / 8, 256, 0, stream>>>(h, att_src, att_dst, a_src, a_dst, N);
    k_edge_alpha<<<(ET + 255) / 256, 256, 0, stream>>>(ei, edge_att, a_src, a_dst,
                                                       kvec, sum_ea, alpha, amax, E, N);
    k_edge_exp<<<(ET + 255) / 256, 256, 0, stream>>>(ei, alpha, amax, denom, E, N);
    k_scatter<<<(ET + 7) / 8, 256, 0, stream>>>(ei, alpha, denom, h, out, E, N);
}